// TaskAlignedAssigner_58531814310479
// MI455X (gfx1250) — compile-verified
//
#include <hip/hip_runtime.h>
#include <cstddef>

// ---------------------------------------------------------------------------
// TaskAlignedAssigner for MI455X (gfx1250, wave32).
// Bandwidth-bound (~500MB traffic => ~25us @ 23.3TB/s). The one-hot einsum
// 'bmc,bac->bma' is a real (64x80)x(80x8400) f32 matmul per batch: mapped to
// V_WMMA_F32_16X16X4_F32 (exact f32, no precision loss), K=80 = 20 steps.
// ---------------------------------------------------------------------------

typedef __attribute__((ext_vector_type(2))) float v2f;
typedef __attribute__((ext_vector_type(8))) float v8f;

#define BS   32
#define MB   64
#define NA   8400
#define NC   80
#define TOPK 13
#define FEPS 1e-9f

// workspace layout (in 4-byte elements); total ~35.2M elems = ~141 MB
constexpr size_t ALIGN_OFF = 0;
constexpr size_t OVL_OFF   = ALIGN_OFF + (size_t)BS*MB*NA;
constexpr size_t TH_OFF    = OVL_OFF   + (size_t)BS*MB*NA;
constexpr size_t PA_OFF    = TH_OFF  + BS*MB;
constexpr size_t PO_OFF    = PA_OFF  + BS*MB;
constexpr size_t TGT_OFF   = PO_OFF  + BS*MB;
constexpr size_t FGW_OFF   = TGT_OFF + (size_t)BS*NA;
constexpr size_t AV_OFF    = FGW_OFF + (size_t)BS*NA;

// ---------------------------------------------------------------------------
// K1: WMMA metric kernel. Block = 128 threads (4 waves), each wave owns one
// 16-anchor N-tile; M=64 GTs -> 4 f32 accumulators (v8f each); K=80 -> 20
// wmma_f32_16x16x4_f32 steps. A (one-hot) synthesized from labels in regs.
// B staged in LDS with row pad 84 (stride 84 mod 64 banks -> 32 lanes hit 32
// distinct banks for the two-dword fragment reads).
// Epilogue: pairwise IoU + iou^6 fused; stores align & overlaps (b,m,a) f32.
// ---------------------------------------------------------------------------
__global__ __launch_bounds__(128) void taa_metric_kernel(
    const float* __restrict__ pd_scores,   // (BS, NA, NC)
    const float* __restrict__ pd_bboxes,   // (BS, NA, 4)
    const int*   __restrict__ gt_labels,   // (BS, MB, 1)
    const float* __restrict__ gt_bboxes,   // (BS, MB, 4)
    float* __restrict__ alignWs,           // (BS, MB, NA)
    float* __restrict__ ovlWs)             // (BS, MB, NA)
{
    __shared__ int   labLds[MB];
    __shared__ float gtLds[MB * 4];
    __shared__ float scLds[4][16 * 84];    // per-wave padded 16x80 score tile

    const int tid  = threadIdx.x;
    const int w    = tid >> 5;
    const int lane = tid & 31;
    const int b    = blockIdx.x / 132;
    const int tg   = blockIdx.x % 132;

    if (tid < MB) labLds[tid] = gt_labels[b * MB + tid];
    for (int i = tid; i < MB * 4; i += 128) gtLds[i] = gt_bboxes[b * MB * 4 + i];
    __syncthreads();

    const int t = tg * 4 + w;              // n-tile id, 525 tiles of 16 anchors
    if (t >= 525) return;

    // stage 16x80 f32 score tile (coalesced float4 loads)
    const float4* src = (const float4*)(pd_scores + ((size_t)b * NA + (size_t)t * 16) * NC);
    for (int i = lane; i < 320; i += 32) {
        float4 v = src[i];
        int row = i / 20, c4 = (i % 20) * 4;
        *(float4*)&scLds[w][row * 84 + c4] = v;
    }
    // same-wave LDS RAW: compiler inserts s_wait_dscnt; no block barrier needed.

    const int n  = lane & 15;              // N (and A-matrix M) index within tile
    const int hi = (lane >> 4) & 1;        // half-wave selects K pair {0,1} vs {2,3}

    int labm[4];
#pragma unroll
    for (int mt = 0; mt < 4; ++mt) labm[mt] = labLds[mt * 16 + n];

    v8f acc[4] = {};                       // C/D: 4 m-tiles of 16x16 f32
    for (int k = 0; k < 20; ++k) {
        const int c0 = 4 * k + 2 * hi;     // class channel for this lane half
        v2f bfrag;                         // B 4x16: VGPR0=K{0|2}, VGPR1=K{1|3}
        bfrag.x = scLds[w][n * 84 + c0];
        bfrag.y = scLds[w][n * 84 + c0 + 1];
#pragma unroll
        for (int mt = 0; mt < 4; ++mt) {
            v2f afrag;                     // one-hot row of A 16x4
            afrag.x = (labm[mt] == c0)     ? 1.0f : 0.0f;
            afrag.y = (labm[mt] == c0 + 1) ? 1.0f : 0.0f;
            acc[mt] = __builtin_amdgcn_wmma_f32_16x16x4_f32(
                false, afrag, false, bfrag, (short)0, acc[mt], false, false);
        }
    }

    // epilogue: fused IoU + align = score * iou^6
    const int a = t * 16 + n;
    const float4 pb = *(const float4*)(pd_bboxes + ((size_t)b * NA + a) * 4);
    const float areaP = fmaxf(pb.z - pb.x, 0.f) * fmaxf(pb.w - pb.y, 0.f);

#pragma unroll
    for (int mt = 0; mt < 4; ++mt) {
#pragma unroll
        for (int j = 0; j < 8; ++j) {
            const int m = mt * 16 + j + hi * 8;   // C/D layout: VGPR j -> M=j / j+8
            const float4 gb = *(const float4*)&gtLds[m * 4];
            float lx = fmaxf(gb.x, pb.x), ly = fmaxf(gb.y, pb.y);
            float rx = fminf(gb.z, pb.z), ry = fminf(gb.w, pb.w);
            float ov = fmaxf(rx - lx, 0.f) * fmaxf(ry - ly, 0.f);
            float areaG = fmaxf(gb.z - gb.x, 0.f) * fmaxf(gb.w - gb.y, 0.f);
            float iou = ov / (areaG + areaP - ov + FEPS);
            float i2 = iou * iou;
            float al = acc[mt][j] * (i2 * i2 * i2);   // ALPHA=1, BETA=6
            size_t o = ((size_t)(b * MB + m)) * NA + a;
            ovlWs[o]   = iou;
            alignWs[o] = al;
        }
    }
}

// ---------------------------------------------------------------------------
// K2: per-(b,m) 13th-largest threshold. One wave per row; each lane keeps a
// register min/max insertion network of 13 (no dynamic indexing => no spill),
// merge 32x13 candidates in LDS, lane 0 selects. Also zero-inits the per-(b,m)
// atomic-max accumulators (required every launch: ws is not re-poisoned).
// ---------------------------------------------------------------------------
__global__ __launch_bounds__(32) void taa_topk_kernel(
    const float* __restrict__ alignWs,
    float* __restrict__ thresh,
    unsigned* __restrict__ posA, unsigned* __restrict__ posO)
{
    __shared__ float cand[32 * TOPK];
    const int r    = blockIdx.x;           // b*MB + m
    const int lane = threadIdx.x;
    const float* row = alignWs + (size_t)r * NA;

    float t[TOPK];
#pragma unroll
    for (int i = 0; i < TOPK; ++i) t[i] = -1.0f;

    for (int a = lane; a < NA; a += 32) {
        float x = row[a];
#pragma unroll
        for (int q = 0; q < TOPK; ++q) {   // sorted-descending insertion
            float hv = fmaxf(t[q], x);
            x = fminf(t[q], x);
            t[q] = hv;
        }
    }
#pragma unroll
    for (int i = 0; i < TOPK; ++i) cand[lane * TOPK + i] = t[i];
    __syncthreads();

    if (lane == 0) {
        float best[TOPK];
#pragma unroll
        for (int i = 0; i < TOPK; ++i) best[i] = -1.0f;
        for (int i = 0; i < 32 * TOPK; ++i) {
            float x = cand[i];
#pragma unroll
            for (int q = 0; q < TOPK; ++q) {
                float hv = fmaxf(best[q], x);
                x = fminf(best[q], x);
                best[q] = hv;
            }
        }
        thresh[r] = best[TOPK - 1];
        posA[r] = 0u;
        posO[r] = 0u;
    }
}

// ---------------------------------------------------------------------------
// K3: thread-per-anchor assignment. Scan m=0..63 (coalesced strided loads):
// positives = (align >= thresh) & in_gt & mask_gt; multi-positive columns
// resolved by argmax IoU over all m (reference semantics). Atomic-max (uint
// bits; all values >= 0) builds pos_align / pos_overlaps per (b,m).
// ---------------------------------------------------------------------------
__global__ __launch_bounds__(256) void taa_assign_kernel(
    const float* __restrict__ anc,         // (NA, 2)
    const float* __restrict__ gtb_g,       // (BS, MB, 4)
    const float* __restrict__ mgt_g,       // (BS, MB, 1)
    const float* __restrict__ alignWs,
    const float* __restrict__ ovlWs,
    const float* __restrict__ thresh,
    unsigned* __restrict__ posA, unsigned* __restrict__ posO,
    int* __restrict__ tgtWs, int* __restrict__ fgWs, float* __restrict__ avalWs)
{
    __shared__ float gtb[MB * 4];
    __shared__ float mg[MB];
    __shared__ float th[MB];

    const int b   = blockIdx.x / 33;
    const int tid = threadIdx.x;
    for (int i = tid; i < MB * 4; i += 256) gtb[i] = gtb_g[b * MB * 4 + i];
    if (tid < MB) { mg[tid] = mgt_g[b * MB + tid]; th[tid] = thresh[b * MB + tid]; }
    __syncthreads();

    const int a = (blockIdx.x % 33) * 256 + tid;
    if (a >= NA) return;

    const float ax = anc[a * 2], ay = anc[a * 2 + 1];
    const size_t base = (size_t)b * MB * NA + a;

    int cnt = 0, tgt1 = 0, bestM = 0;
    float a1v = 0.f, o1v = 0.f, bestOv = -1.f, bestAl = 0.f;

    for (int m = 0; m < MB; ++m) {
        float al = alignWs[base + (size_t)m * NA];
        float ov = ovlWs[base + (size_t)m * NA];
        float gx1 = gtb[m * 4], gy1 = gtb[m * 4 + 1];
        float gx2 = gtb[m * 4 + 2], gy2 = gtb[m * 4 + 3];
        float dmin = fminf(fminf(ax - gx1, ay - gy1), fminf(gx2 - ax, gy2 - ay));
        bool pos = (al >= th[m]) && (dmin > FEPS) && (mg[m] > 0.f);
        if (pos) { if (cnt == 0) { tgt1 = m; a1v = al; o1v = ov; } cnt++; }
        if (ov > bestOv) { bestOv = ov; bestM = m; bestAl = al; }  // first-max on ties
    }

    int fg, tgt; float aval, oval;
    if (cnt > 1)       { fg = 1; tgt = bestM; aval = bestAl; oval = bestOv; }
    else if (cnt == 1) { fg = 1; tgt = tgt1;  aval = a1v;    oval = o1v;    }
    else               { fg = 0; tgt = 0;     aval = 0.f;    oval = 0.f;    }

    const size_t g = (size_t)b * NA + a;
    tgtWs[g]  = tgt;
    fgWs[g]   = fg;
    avalWs[g] = aval;
    if (fg) {
        atomicMax(&posA[b * MB + tgt], __float_as_uint(aval));
        atomicMax(&posO[b * MB + tgt], __float_as_uint(oval));
    }
}

// ---------------------------------------------------------------------------
// K4: finalize. norm = align * pos_ovl / (pos_align + eps); labels/boxes
// gathered via target idx; 86MB target_scores written with a block-cooperative
// fully-coalesced one-hot scatter (per-anchor label/norm staged in LDS).
// ---------------------------------------------------------------------------
__global__ __launch_bounds__(256) void taa_finalize_kernel(
    const int*   __restrict__ gt_labels,
    const float* __restrict__ gt_bboxes,
    const unsigned* __restrict__ posA, const unsigned* __restrict__ posO,
    const int* __restrict__ tgtWs, const int* __restrict__ fgWs,
    const float* __restrict__ avalWs,
    int* __restrict__ outLab, float* __restrict__ outBox,
    float* __restrict__ outSco, float* __restrict__ outFg)
{
    __shared__ int   tlL[256];
    __shared__ float nmL[256];

    const int b   = blockIdx.x / 33;
    const int tid = threadIdx.x;
    const int a0  = (blockIdx.x % 33) * 256;
    const int a   = a0 + tid;

    int tl = NC - 1; float nm = 0.f;
    if (a < NA) {
        const size_t g = (size_t)b * NA + a;
        const int tgt = tgtWs[g];
        const int fg  = fgWs[g];
        const float pa = __uint_as_float(posA[b * MB + tgt]);
        const float po = __uint_as_float(posO[b * MB + tgt]);
        nm = fg ? avalWs[g] * po / (pa + FEPS) : 0.f;
        const int lab = gt_labels[b * MB + tgt];
        tl = fg ? min(max(lab, 0), NC - 1) : NC - 1;   // !fg -> clip(C)=79
        outLab[g] = tl;
        const float4 gb = *(const float4*)(gt_bboxes + (size_t)(b * MB + tgt) * 4);
        *(float4*)(outBox + g * 4) = gb;               // gathered even when !fg (tgt=0)
        outFg[g] = fg ? 1.0f : 0.0f;
    }
    tlL[tid] = tl;
    nmL[tid] = nm;
    __syncthreads();

    const size_t sbase = ((size_t)b * NA + a0) * NC;
    const int nAnch = min(256, NA - a0);
    for (int i = tid; i < nAnch * NC; i += 256) {
        int la = i / NC, c = i % NC;
        outSco[sbase + i] = (c == tlL[la]) ? nmL[la] : 0.f;   // onehot*fg*norm
    }
}

// ---------------------------------------------------------------------------
extern "C" void kernel_launch(void* const* d_in, const int* in_sizes, int n_in,
                              void* d_out, int out_size, void* d_ws, size_t ws_size,
                              hipStream_t stream)
{
    const float* pd_scores = (const float*)d_in[0];
    const float* pd_bboxes = (const float*)d_in[1];
    const float* anc       = (const float*)d_in[2];
    const int*   gt_labels = (const int*)d_in[3];
    const float* gt_bboxes = (const float*)d_in[4];
    const float* mask_gt   = (const float*)d_in[5];

    float*    ws      = (float*)d_ws;
    float*    alignWs = ws + ALIGN_OFF;
    float*    ovlWs   = ws + OVL_OFF;
    float*    thresh  = ws + TH_OFF;
    unsigned* posA    = (unsigned*)(ws + PA_OFF);
    unsigned* posO    = (unsigned*)(ws + PO_OFF);
    int*      tgtWs   = (int*)(ws + TGT_OFF);
    int*      fgWs    = (int*)(ws + FGW_OFF);
    float*    avalWs  = ws + AV_OFF;

    // outputs concatenated in return order: labels(i32), bboxes, scores, fg
    int*   outLab = (int*)d_out;
    float* outBox = (float*)d_out + (size_t)BS * NA;
    float* outSco = outBox + (size_t)BS * NA * 4;
    float* outFg  = outSco + (size_t)BS * NA * NC;

    taa_metric_kernel<<<dim3(BS * 132), 128, 0, stream>>>(
        pd_scores, pd_bboxes, gt_labels, gt_bboxes, alignWs, ovlWs);

    taa_topk_kernel<<<dim3(BS * MB), 32, 0, stream>>>(
        alignWs, thresh, posA, posO);

    taa_assign_kernel<<<dim3(BS * 33), 256, 0, stream>>>(
        anc, gt_bboxes, mask_gt, alignWs, ovlWs, thresh,
        posA, posO, tgtWs, fgWs, avalWs);

    taa_finalize_kernel<<<dim3(BS * 33), 256, 0, stream>>>(
        gt_labels, gt_bboxes, posA, posO, tgtWs, fgWs, avalWs,
        outLab, outBox, outSco, outFg);
}